// Cheb_Conv_41721312314188
// MI455X (gfx1250) — compile-verified
//
#include <hip/hip_runtime.h>
#include <math.h>

typedef __attribute__((ext_vector_type(2))) float v2f;
typedef __attribute__((ext_vector_type(8))) float v8f;

// Global (addrspace(1)) pointer aliases: force A-fragment loads in the GEMM to
// lower to global_load_b64 (LOADcnt only) instead of flat_load_b64, which on
// CDNA5 occupies the LDS path (DScnt) and contends with the B-fragment ds_loads.
typedef const float __attribute__((address_space(1))) gcfloat;
typedef const v2f   __attribute__((address_space(1))) gcv2f;

// ---------------------------------------------------------------------------
// ChebConv K=4, F_IN=F_OUT=128 on gfx1250 (MI455X).
// SpMM passes are L2-resident (node features = 51MB < 192MB L2); the dense
// 100000x512 @ 512x128 GEMM runs on the f32 WMMA path (v_wmma_f32_16x16x4_f32)
// with W staged in LDS (64KB, bank-conflict-free layout).
// ---------------------------------------------------------------------------

__global__ __launch_bounds__(256) void fill_zero_f4(float4* __restrict__ p, long n4) {
    long i = (long)blockIdx.x * 256 + threadIdx.x;
    if (i < n4) p[i] = make_float4(0.f, 0.f, 0.f, 0.f);
}

__global__ __launch_bounds__(256) void fill_zero_f1(float* __restrict__ p, int n) {
    int i = blockIdx.x * 256 + threadIdx.x;
    if (i < n) p[i] = 0.f;
}

__global__ __launch_bounds__(256) void degree_kernel(const int* __restrict__ dst,
                                                     float* __restrict__ deg, int E) {
    int i = blockIdx.x * 256 + threadIdx.x;
    if (i < E) atomicAdd(&deg[dst[i]], 1.0f);
}

__global__ __launch_bounds__(256) void dsqrt_kernel(const float* __restrict__ deg,
                                                    float* __restrict__ ds, int N) {
    int i = blockIdx.x * 256 + threadIdx.x;
    if (i < N) ds[i] = 1.0f / sqrtf(fmaxf(deg[i], 1.0f));
}

// One wave32 per edge; each lane owns 4 of the 128 features (float4 slice).
// agg[dst] += x[src] * ds[src]
__global__ __launch_bounds__(256) void edge_scatter(const float* __restrict__ X,
                                                    const int* __restrict__ src,
                                                    const int* __restrict__ dst,
                                                    const float* __restrict__ ds,
                                                    float* __restrict__ agg, int E) {
    int gw = (blockIdx.x * 256 + threadIdx.x) >> 5;
    int lane = threadIdx.x & 31;
    if (gw >= E) return;
    int s = __builtin_amdgcn_readfirstlane(src[gw]);
    int d = __builtin_amdgcn_readfirstlane(dst[gw]);
    float scale = ds[s];
    const float4 v = *(const float4*)(X + (size_t)s * 128 + lane * 4);
    float* o = agg + (size_t)d * 128 + lane * 4;
    atomicAdd(o + 0, v.x * scale);
    atomicAdd(o + 1, v.y * scale);
    atomicAdd(o + 2, v.z * scale);
    atomicAdd(o + 3, v.w * scale);
}

// In place: buf = -r*(buf*ds[node]) + (r-1)*X0     (X1 formula), r = 2/lambda_max
__global__ __launch_bounds__(256) void combine_first(float* __restrict__ buf,
                                                     const float* __restrict__ X0,
                                                     const float* __restrict__ ds,
                                                     const float* __restrict__ lm, long n4) {
    long i = (long)blockIdx.x * 256 + threadIdx.x;
    if (i >= n4) return;
    float r = 2.0f / lm[0];
    int node = (int)(i >> 5);
    float c0 = -r * ds[node];
    float c1 = r - 1.0f;
    float4 h = ((const float4*)buf)[i];
    float4 x0 = ((const float4*)X0)[i];
    float4 o;
    o.x = c0 * h.x + c1 * x0.x;
    o.y = c0 * h.y + c1 * x0.y;
    o.z = c0 * h.z + c1 * x0.z;
    o.w = c0 * h.w + c1 * x0.w;
    ((float4*)buf)[i] = o;
}

// In place: buf = -2r*(buf*ds[node]) + 2(r-1)*Xp - Xpp
__global__ __launch_bounds__(256) void combine_next(float* __restrict__ buf,
                                                    const float* __restrict__ Xp,
                                                    const float* __restrict__ Xpp,
                                                    const float* __restrict__ ds,
                                                    const float* __restrict__ lm, long n4) {
    long i = (long)blockIdx.x * 256 + threadIdx.x;
    if (i >= n4) return;
    float r = 2.0f / lm[0];
    int node = (int)(i >> 5);
    float c0 = -2.0f * r * ds[node];
    float c1 = 2.0f * (r - 1.0f);
    float4 h = ((const float4*)buf)[i];
    float4 xp = ((const float4*)Xp)[i];
    float4 xq = ((const float4*)Xpp)[i];
    float4 o;
    o.x = c0 * h.x + c1 * xp.x - xq.x;
    o.y = c0 * h.y + c1 * xp.y - xq.y;
    o.z = c0 * h.z + c1 * xp.z - xq.z;
    o.w = c0 * h.w + c1 * xp.w - xq.w;
    ((float4*)buf)[i] = o;
}

// out[n, o] = b[o] + sum_a sum_k Xa[n,k] * W[o, a*128+k]
// One wave computes a 16-node x 128-col tile via v_wmma_f32_16x16x4_f32.
// W block (128x128) for the current Chebyshev order is staged in LDS with a
// k-quad-interleaved layout: sW[(k>>2)*512 + o*4 + (k&3)]  -> conflict-free
// 8B ds loads (banks 4*row and 4*row+2 across the wave).
__global__ __launch_bounds__(256) void cheb_gemm_wmma(const float* __restrict__ X0,
                                                      const float* __restrict__ X1,
                                                      const float* __restrict__ X2,
                                                      const float* __restrict__ X3,
                                                      const float* __restrict__ W,
                                                      const float* __restrict__ bias,
                                                      float* __restrict__ out,
                                                      int N, int ntiles) {
    __shared__ float sW[128 * 128];  // 64 KB
    const int lane = threadIdx.x & 31;
    const int wv   = threadIdx.x >> 5;
    const int tile = blockIdx.x * 8 + wv;
    const bool active = tile < ntiles;
    const int row = lane & 15;     // M row (A) or N col (B/D) within the 16x16 tile
    const int kh  = lane >> 4;     // K half select: +0 or +2
    int node = (active ? tile : 0) * 16 + row;
    if (node >= N) node = N - 1;   // clamp for partial last tile (loads only)

    const float* Xs[4] = {X0, X1, X2, X3};
    v8f zero = {0.f, 0.f, 0.f, 0.f, 0.f, 0.f, 0.f, 0.f};
    v8f acc[8];
#pragma unroll
    for (int t = 0; t < 8; ++t) acc[t] = zero;

#pragma unroll 1
    for (int a = 0; a < 4; ++a) {
        __syncthreads();
        // stage W[:, a*128 .. a*128+127] into LDS (4096 float4 across 256 threads)
#pragma unroll
        for (int it = 0; it < 16; ++it) {
            int idx = it * 256 + threadIdx.x;   // 0..4095
            int o   = idx >> 5;                 // output row 0..127
            int kq  = (idx & 31) << 2;          // k quad base 0..124
            float4 w4 = *(const float4*)(W + (size_t)o * 512 + a * 128 + kq);
            *(float4*)(&sW[(kq >> 2) * 512 + o * 4]) = w4;
        }
        __syncthreads();

        // Explicit addrspace(1) pointer: guarantees global_load_b64 for the A
        // fragment (keeps the LDS pipe free for the B-fragment ds_loads).
        gcfloat* rowA = (gcfloat*)(Xs[a] + (size_t)node * 128 + 2 * kh);
#pragma unroll 4
        for (int kk = 0; kk < 128; kk += 4) {
            // A fragment: 16x4 f32 tile; VGPR0={K0 | K2}, VGPR1={K1 | K3}
            v2f av = *(gcv2f*)(rowA + kk);
            const float* sB = &sW[(kk >> 2) * 512 + 2 * kh];
            // Preload all 8 B fragments into distinct registers so the four
            // ds_load_2addr_b64 issue back-to-back with ONE dscnt wait, then
            // the 8 WMMAs run as an uninterrupted burst.
            v2f bv[8];
#pragma unroll
            for (int t = 0; t < 8; ++t) {
                bv[t] = *(const v2f*)(sB + (t * 16 + row) * 4);
            }
#pragma unroll
            for (int t = 0; t < 8; ++t) {
                acc[t] = __builtin_amdgcn_wmma_f32_16x16x4_f32(
                    false, av, false, bv[t], (short)0, acc[t], false, false);
            }
        }
    }

    if (!active) return;
    const int nbase = tile * 16;
    // D layout: VGPR j -> M=j (lanes 0-15), M=j+8 (lanes 16-31); N = col = lane&15
#pragma unroll
    for (int t = 0; t < 8; ++t) {
        const int col = t * 16 + row;
        const float bb = bias[col];
#pragma unroll
        for (int j = 0; j < 8; ++j) {
            const int m = nbase + j + 8 * kh;
            if (m < N) out[(size_t)m * 128 + col] = acc[t][j] + bb;
        }
    }
}

extern "C" void kernel_launch(void* const* d_in, const int* in_sizes, int n_in,
                              void* d_out, int out_size, void* d_ws, size_t ws_size,
                              hipStream_t stream) {
    const float* signal = (const float*)d_in[0];  // N x 128
    const int*   src    = (const int*)d_in[1];    // E
    const int*   dst    = (const int*)d_in[2];    // E
    const float* W      = (const float*)d_in[3];  // 128 x 512
    const float* bias   = (const float*)d_in[4];  // 128
    const float* lm     = (const float*)d_in[5];  // 1

    const int N = in_sizes[0] / 128;
    const int E = in_sizes[1];

    float* ws  = (float*)d_ws;
    float* deg = ws;
    float* ds  = ws + N;
    float* X1  = ws + 2 * (size_t)N;
    float* X2  = X1 + (size_t)N * 128;
    float* X3  = X2 + (size_t)N * 128;
    float* out = (float*)d_out;

    const long n4 = (long)N * 32;                       // N*128 floats as float4
    const unsigned gN    = (unsigned)((N + 255) / 256);
    const unsigned gE    = (unsigned)((E + 255) / 256);
    const unsigned gFeat = (unsigned)((n4 + 255) / 256);
    const unsigned gEdge = (unsigned)((E + 7) / 8);     // one wave32 per edge
    const int ntiles = (N + 15) / 16;
    const unsigned gGemm = (unsigned)((ntiles + 7) / 8);

    // degree and d^-1/2
    fill_zero_f1<<<gN, 256, 0, stream>>>(deg, N);
    degree_kernel<<<gE, 256, 0, stream>>>(dst, deg, E);
    dsqrt_kernel<<<gN, 256, 0, stream>>>(deg, ds, N);

    // X1 = -r*L(X0) + (r-1)*X0
    fill_zero_f4<<<gFeat, 256, 0, stream>>>((float4*)X1, n4);
    edge_scatter<<<gEdge, 256, 0, stream>>>(signal, src, dst, ds, X1, E);
    combine_first<<<gFeat, 256, 0, stream>>>(X1, signal, ds, lm, n4);

    // X2 = -2r*L(X1) + 2(r-1)*X1 - X0
    fill_zero_f4<<<gFeat, 256, 0, stream>>>((float4*)X2, n4);
    edge_scatter<<<gEdge, 256, 0, stream>>>(X1, src, dst, ds, X2, E);
    combine_next<<<gFeat, 256, 0, stream>>>(X2, X1, signal, ds, lm, n4);

    // X3 = -2r*L(X2) + 2(r-1)*X2 - X1
    fill_zero_f4<<<gFeat, 256, 0, stream>>>((float4*)X3, n4);
    edge_scatter<<<gEdge, 256, 0, stream>>>(X2, src, dst, ds, X3, E);
    combine_next<<<gFeat, 256, 0, stream>>>(X3, X2, X1, ds, lm, n4);

    // out = [X0|X1|X2|X3] @ W.T + b   (WMMA f32)
    cheb_gemm_wmma<<<gGemm, 256, 0, stream>>>(signal, X1, X2, X3, W, bias, out, N, ntiles);
}